// ProteinFeaturesLigand_60017872994851
// MI455X (gfx1250) — compile-verified
//
#include <hip/hip_runtime.h>
#include <stdint.h>

#define NUM_RBF   16
#define K_NEIGH   48
#define POS_DIM   16
#define EDGE_OUT  128
#define N_ATOMS   14
#define L_SEQ     2048
#define NEDGE_IN  416      // 16 pos + 400 rbf
#define FSTRIDE   420      // padded LDS row stride (mod 64 = 36, 16-way conflict free)
#define TILE_M    32       // edges per block (2 x 16 M-tiles)
#define ESTRIDE   132      // padded stride for pre-LN output tile

typedef float v2f __attribute__((ext_vector_type(2)));
typedef float v8f __attribute__((ext_vector_type(8)));

__device__ __forceinline__ unsigned long long umin64(unsigned long long a, unsigned long long b) {
    return a < b ? a : b;
}

// ---------------- Kernel 1: build X_all = [4 gathered backbone atoms, virtual atom] ----------------
__global__ void build_xall(const float* __restrict__ X, const int* __restrict__ S,
                           const int* __restrict__ table, float* __restrict__ Xall, int NR) {
    int i = blockIdx.x * blockDim.x + threadIdx.x;
    if (i >= NR) return;
    int s = S[i];
    const float* xi = X + (size_t)i * N_ATOMS * 3;
    float bb[4][3];
    #pragma unroll
    for (int t = 0; t < 4; ++t) {
        int a = table[s * 4 + t];
        bb[t][0] = xi[a * 3 + 0];
        bb[t][1] = xi[a * 3 + 1];
        bb[t][2] = xi[a * 3 + 2];
    }
    float b1[3], b2[3], nr[3];
    #pragma unroll
    for (int c = 0; c < 3; ++c) { b1[c] = bb[0][c] - bb[1][c]; b2[c] = bb[2][c] - bb[1][c]; }
    nr[0] = b1[1] * b2[2] - b1[2] * b2[1];
    nr[1] = b1[2] * b2[0] - b1[0] * b2[2];
    nr[2] = b1[0] * b2[1] - b1[1] * b2[0];
    const float WN = 0.58273431f, WB1 = -0.56802827f, WB2 = -0.54067466f;
    float* o = Xall + (size_t)i * 15;
    #pragma unroll
    for (int t = 0; t < 4; ++t)
        #pragma unroll
        for (int c = 0; c < 3; ++c) o[t * 3 + c] = bb[t][c];
    #pragma unroll
    for (int c = 0; c < 3; ++c) o[12 + c] = WN * nr[c] + WB1 * b1[c] + WB2 * b2[c] + bb[1][c];
}

// ---------------- Kernel 2: K-NN (48 smallest CA-CA distances, lowest-index tiebreak) ----------------
__global__ void __launch_bounds__(256)
knn_topk(const float* __restrict__ Xall, const float* __restrict__ mask,
         int* __restrict__ EidxWs, int* __restrict__ EidxOut) {
    __shared__ unsigned long long keys[L_SEQ];
    __shared__ unsigned long long red[256];
    int i   = blockIdx.x;           // flat residue index b*L + l
    int b   = i / L_SEQ;
    int tid = threadIdx.x;
    float cax = Xall[(size_t)i * 15 + 3];
    float cay = Xall[(size_t)i * 15 + 4];
    float caz = Xall[(size_t)i * 15 + 5];
    float mi  = mask[i];
    for (int j = tid; j < L_SEQ; j += 256) {
        size_t jj = (size_t)(b * L_SEQ + j) * 15;
        float dx = cax - Xall[jj + 3];
        float dy = cay - Xall[jj + 4];
        float dz = caz - Xall[jj + 5];
        float D  = sqrtf(dx * dx + dy * dy + dz * dz + 1e-6f);
        float m2 = mi * mask[b * L_SEQ + j];
        float Da = D * m2 + (1.0f - m2) * 1000000.0f;   // non-negative -> bit-monotonic
        keys[j] = ((unsigned long long)__float_as_uint(Da) << 32) | (unsigned)j;
    }
    __syncthreads();
    for (int it = 0; it < K_NEIGH; ++it) {
        unsigned long long best = ~0ull;
        for (int j = tid; j < L_SEQ; j += 256) best = umin64(best, keys[j]);
        red[tid] = best;
        __syncthreads();
        #pragma unroll
        for (int off = 128; off > 0; off >>= 1) {
            if (tid < off) red[tid] = umin64(red[tid], red[tid + off]);
            __syncthreads();
        }
        if (tid == 0) {
            int jw = (int)(red[0] & 0xFFFFFFFFull);
            EidxWs[(size_t)i * K_NEIGH + it]  = jw;
            EidxOut[(size_t)i * K_NEIGH + it] = jw;
            keys[jw] = ~0ull;                 // remove winner
        }
        __syncthreads();
    }
}

// ---------------- Kernel 3: per-edge features -> WMMA f32 GEMM (416 -> 128) -> LayerNorm ----------------
// 32 edges per block; each wave owns one 16-wide N-tile and TWO 16-row M-tiles so every
// W_e (B-matrix) fragment load is reused twice. El overlays F (barrier-separated phases).
__global__ void __launch_bounds__(256)
edge_gemm(const float* __restrict__ Xall, const int* __restrict__ Eidx,
          const int* __restrict__ ridx, const float* __restrict__ Wpos,
          const float* __restrict__ bpos, const float* __restrict__ We,
          const float* __restrict__ lng, const float* __restrict__ lnb,
          float* __restrict__ out) {
    __shared__ float smem[TILE_M * FSTRIDE];  // features F (32 x 416 padded); later re-used as El
    __shared__ float Dl[TILE_M * 25];         // 25 atom-pair distances per edge
    __shared__ float r1[256], r2[256];
    float* F  = smem;
    float* El = smem;                          // overlay: 32 x 132 = 16.9KB < 53.8KB
    int tid = threadIdx.x;
    int e0  = blockIdx.x * TILE_M;

    // --- phase 1: atom-pair distances ---
    for (int p = tid; p < TILE_M * 25; p += 256) {
        int m = p / 25, pr = p % 25;
        int a = pr / 5, bt = pr % 5;
        int e = e0 + m;
        int iflat = e / K_NEIGH;
        int b = iflat / L_SEQ;
        int j = Eidx[e];
        const float* qa = Xall + (size_t)iflat * 15 + a * 3;
        const float* nb = Xall + (size_t)(b * L_SEQ + j) * 15 + bt * 3;
        float dx = qa[0] - nb[0], dy = qa[1] - nb[1], dz = qa[2] - nb[2];
        Dl[p] = sqrtf(dx * dx + dy * dy + dz * dz + 1e-6f);
    }
    // --- positional features (one-hot @ W_pos == row gather) ---
    for (int p = tid; p < TILE_M * POS_DIM; p += 256) {
        int m = p / POS_DIM, c = p % POS_DIM;
        int e = e0 + m;
        int iflat = e / K_NEIGH;
        int b = iflat / L_SEQ;
        int j = Eidx[e];
        int off = ridx[iflat] - ridx[b * L_SEQ + j] + 32;
        off = min(max(off, 0), 64);
        F[m * FSTRIDE + c] = Wpos[off * POS_DIM + c] + bpos[c];
    }
    __syncthreads();
    // --- phase 2: RBF expansion (feature col = 16 + a*80 + bt*16 + r) ---
    for (int p = tid; p < TILE_M * 400; p += 256) {
        int m = p / 400, c = p % 400;
        int pr = c >> 4, r = c & 15;
        float D  = Dl[m * 25 + pr];
        float mu = 2.0f + (float)r * (20.0f / 15.0f);
        float t  = (D - mu) * 0.8f;            // sigma = 1.25
        F[m * FSTRIDE + 16 + c] = __expf(-t * t);
    }
    __syncthreads();

    // --- GEMM: wave w -> N-tile n0 = 16w; two M-tiles share each B fragment. K = 416 = 104*4 ---
    int wave = tid >> 5;
    int lane = tid & 31;
    int half = lane >> 4;       // lane-half selects K pair {0,1} vs {2,3}
    int lq   = lane & 15;
    int n0   = wave * 16;
    v8f acc0 = {}, acc1 = {};
    for (int kk = 0; kk < 104; ++kk) {
        int k0 = kk * 4;
        // B 4x16 f32: lanes 0-15 -> rows {K0,K1}, lanes 16-31 -> rows {K2,K3}, N = lq
        v2f bf;
        bf.x = We[(size_t)(k0 + 2 * half) * EDGE_OUT + n0 + lq];
        bf.y = We[(size_t)(k0 + 2 * half + 1) * EDGE_OUT + n0 + lq];
        // A 16x4 f32: lanes 0-15 -> {K0,K1}, lanes 16-31 -> {K2,K3}, M = lq
        v2f a0, a1;
        a0.x = F[lq * FSTRIDE + k0 + 2 * half];
        a0.y = F[lq * FSTRIDE + k0 + 2 * half + 1];
        a1.x = F[(16 + lq) * FSTRIDE + k0 + 2 * half];
        a1.y = F[(16 + lq) * FSTRIDE + k0 + 2 * half + 1];
        acc0 = __builtin_amdgcn_wmma_f32_16x16x4_f32(false, a0, false, bf,
                                                     (short)0, acc0, false, false);
        acc1 = __builtin_amdgcn_wmma_f32_16x16x4_f32(false, a1, false, bf,
                                                     (short)0, acc1, false, false);
    }
    __syncthreads();   // all waves done reading F -> safe to overlay El on it

    // C/D layout: VGPR j -> M = j (lanes 0-15) or j+8 (lanes 16-31), N = lq
    #pragma unroll
    for (int j = 0; j < 8; ++j) {
        int row = j + half * 8;
        El[row * ESTRIDE + n0 + lq]        = acc0[j];
        El[(16 + row) * ESTRIDE + n0 + lq] = acc1[j];
    }
    __syncthreads();

    // --- LayerNorm over 128 outputs per edge (8 threads / edge, 16 values each) ---
    int m   = tid >> 3;
    int sub = tid & 7;
    float s1 = 0.f, s2 = 0.f;
    #pragma unroll
    for (int q = 0; q < 16; ++q) {
        float v = El[m * ESTRIDE + sub * 16 + q];
        s1 += v; s2 += v * v;
    }
    r1[tid] = s1; r2[tid] = s2;
    __syncthreads();
    #pragma unroll
    for (int off = 4; off > 0; off >>= 1) {
        if (sub < off) { r1[tid] += r1[tid + off]; r2[tid] += r2[tid + off]; }
        __syncthreads();
    }
    float S1 = r1[tid & ~7], S2 = r2[tid & ~7];
    float mean = S1 * (1.0f / 128.0f);
    float var  = S2 * (1.0f / 128.0f) - mean * mean;   // population variance (jnp.var)
    float rs   = rsqrtf(var + 1e-5f);
    size_t base = ((size_t)e0 + m) * EDGE_OUT;
    #pragma unroll
    for (int q = 0; q < 16; ++q) {
        int n = sub * 16 + q;
        float v = El[m * ESTRIDE + n];
        out[base + n] = (v - mean) * rs * lng[n] + lnb[n];
    }
}

extern "C" void kernel_launch(void* const* d_in, const int* in_sizes, int n_in,
                              void* d_out, int out_size, void* d_ws, size_t ws_size,
                              hipStream_t stream) {
    const float* X     = (const float*)d_in[0];
    // d_in[1] = X_m (unused by the reference)
    const int*   S     = (const int*)d_in[2];
    const int*   ridx  = (const int*)d_in[3];
    const float* mask  = (const float*)d_in[4];
    const int*   table = (const int*)d_in[5];
    const float* Wpos  = (const float*)d_in[6];
    const float* bpos  = (const float*)d_in[7];
    const float* We    = (const float*)d_in[8];
    const float* lng   = (const float*)d_in[9];
    const float* lnb   = (const float*)d_in[10];

    const int NR = in_sizes[2];              // B*L = 4096
    float* Xall   = (float*)d_ws;            // NR*15 floats
    int*   EidxWs = (int*)((char*)d_ws + (size_t)NR * 15 * sizeof(float));
    float* outE   = (float*)d_out;           // NR*48*128 floats
    int*   EidxOut = (int*)((float*)d_out + (size_t)NR * K_NEIGH * EDGE_OUT);

    build_xall<<<(NR + 63) / 64, 64, 0, stream>>>(X, S, table, Xall, NR);
    knn_topk<<<NR, 256, 0, stream>>>(Xall, mask, EidxWs, EidxOut);
    const int nedges = NR * K_NEIGH;         // 196,608
    edge_gemm<<<nedges / TILE_M, 256, 0, stream>>>(Xall, EidxWs, ridx, Wpos, bpos,
                                                   We, lng, lnb, outE);
}